// BipartiteGraph_86346022518992
// MI455X (gfx1250) — compile-verified
//
#include <hip/hip_runtime.h>

typedef __attribute__((ext_vector_type(2))) float v2f;
typedef __attribute__((ext_vector_type(8))) float v8f;

#define K_TOT   6144
#define M_TOT   6144
#define KSPLIT  8
#define KCHUNK  (K_TOT / KSPLIT)     // 768 = 4 * 192 (192 = scale period = 48 k-steps)
#define GSTEPS  12                   // k-steps (of 4 k) per pipeline group
#define NSTEPS  (KCHUNK / 4)         // 192 steps per wave
#define NGROUP  (NSTEPS / GSTEPS)    // 16 groups (even -> clean 2-stage pipeline)

template<bool NT>
__device__ __forceinline__ float ldW(const float* p) {
    if (NT) return __builtin_nontemporal_load(p);
    return *p;
}

// D[16 x 6144] = A[16 x 6144] x (W[6144 x 6144] .* scale)
// scale(k,m) = scl[k % 192][m >> 5] -> per-tile 192-entry LDS table folded into A frag.
// DUAL: two GEMMs share the W and A streams with different scale tables.
// NT:   stream W non-temporally (pass 2) so s2t stays L2-resident for pass 3.
// Software-pipelined: register double buffer, ~1.5 groups (36+ loads) in flight.
template<bool DUAL, bool NT>
__global__ __launch_bounds__(256) void gemm_scaled(
    const float* __restrict__ Ain, const float* __restrict__ W,
    const float* __restrict__ sclA, const float* __restrict__ sclB,
    float* __restrict__ outA, float* __restrict__ outB)
{
    __shared__ float scl[2][192];
    __shared__ float redA[KSPLIT][256];
    __shared__ float redB[KSPLIT][256];

    const int tid  = threadIdx.x;
    const int lane = tid & 31;
    const int w    = tid >> 5;           // K-split wave id, 0..7
    const int tile = blockIdx.x;         // 0..383
    const int m0   = tile * 16;
    const int o    = tile >> 1;          // m0 >> 5, constant per 16-wide tile

    if (tid < 192) {
        scl[0][tid] = sclA[tid * 192 + o];
        if (DUAL) scl[1][tid] = sclB[tid * 192 + o];
    }
    __syncthreads();

    // A 16x4 f32 frag: lanes 0-15 -> M=lane, V0:K=k0, V1:K=k0+1;
    //                  lanes 16-31 -> M=lane-16, K=k0+2 / k0+3.
    // B 4x16 f32 frag (symmetric): lane -> N=lane&15, same half-split K offsets.
    const int n    = lane & 15;
    const int hi2  = (lane >> 4) << 1;
    const int kstart = w * KCHUNK;       // KCHUNK % 192 == 0 -> scale phase = hi2
    const float* aPtr = Ain + n * K_TOT + kstart + hi2;
    const float* bPtr = W + (size_t)(kstart + hi2) * M_TOT + m0 + n;
    const float* sPtrA = &scl[0][hi2];
    const float* sPtrB = &scl[1][hi2];

    v8f accA = {0.f,0.f,0.f,0.f,0.f,0.f,0.f,0.f};
    v8f accB = {0.f,0.f,0.f,0.f,0.f,0.f,0.f,0.f};

    float ax0[GSTEPS], ay0[GSTEPS], bx0[GSTEPS], by0[GSTEPS];
    float ax1[GSTEPS], ay1[GSTEPS], bx1[GSTEPS], by1[GSTEPS];

    auto loadG = [&](int g, float* ax, float* ay, float* bx, float* by) {
        const float* ap = aPtr + g * GSTEPS * 4;
        const float* bp = bPtr + (size_t)(g * GSTEPS * 4) * M_TOT;
        #pragma unroll
        for (int t = 0; t < GSTEPS; ++t) {
            float2 a = *(const float2*)(ap + t * 4);
            ax[t] = a.x; ay[t] = a.y;
            bx[t] = ldW<NT>(bp + (size_t)(t * 4 + 0) * M_TOT);
            by[t] = ldW<NT>(bp + (size_t)(t * 4 + 1) * M_TOT);
        }
    };
    auto compG = [&](int g, const float* ax, const float* ay,
                     const float* bx, const float* by) {
        const int sBase = (g & 3) * (GSTEPS * 4);    // 48-step scale period
        #pragma unroll
        for (int t = 0; t < GSTEPS; ++t) {
            v2f bF; bF.x = bx[t]; bF.y = by[t];
            float2 sA = *(const float2*)(sPtrA + sBase + t * 4);
            v2f aA; aA.x = ax[t] * sA.x; aA.y = ay[t] * sA.y;
            accA = __builtin_amdgcn_wmma_f32_16x16x4_f32(
                false, aA, false, bF, (short)0, accA, false, false);
            if (DUAL) {
                float2 sB = *(const float2*)(sPtrB + sBase + t * 4);
                v2f aB; aB.x = ax[t] * sB.x; aB.y = ay[t] * sB.y;
                accB = __builtin_amdgcn_wmma_f32_16x16x4_f32(
                    false, aB, false, bF, (short)0, accB, false, false);
            }
        }
    };

    // Two-stage software pipeline: while computing group g, group g+1's loads
    // are in flight and group g+2's are issued right after g is consumed.
    loadG(0, ax0, ay0, bx0, by0);
    loadG(1, ax1, ay1, bx1, by1);
    for (int g = 0; g + 2 < NGROUP; g += 2) {
        compG(g, ax0, ay0, bx0, by0);
        loadG(g + 2, ax0, ay0, bx0, by0);
        compG(g + 1, ax1, ay1, bx1, by1);
        if (g + 3 < NGROUP)
            loadG(g + 3, ax1, ay1, bx1, by1);
    }
    compG(NGROUP - 2, ax0, ay0, bx0, by0);
    compG(NGROUP - 1, ax1, ay1, bx1, by1);

    // Cross-wave (K-split) reduction via LDS — deterministic, no FP atomics.
    #pragma unroll
    for (int v = 0; v < 8; ++v) {
        redA[w][lane * 8 + v] = accA[v];
        if (DUAL) redB[w][lane * 8 + v] = accB[v];
    }
    __syncthreads();

    // 256 D-tile entries; each thread finalizes one.
    // C/D layout: p = lane*8+v  ->  n = v + 8*(lane>=16), col = lane&15.
    {
        int p  = tid;
        int lp = p >> 3, v = p & 7;
        int nn  = v + ((lp >> 4) << 3);
        int col = lp & 15;
        int mc  = m0 + col;
        int oi  = nn * M_TOT + (mc & 31) * 192 + (mc >> 5);
        float sA = 0.f, sB = 0.f;
        #pragma unroll
        for (int ww = 0; ww < KSPLIT; ++ww) {
            sA += redA[ww][p];
            if (DUAL) sB += redB[ww][p];
        }
        outA[oi] = sA;
        if (DUAL) outB[oi] = sB;
    }
}

// out[512 x 192] = X[512 x 192] @ Mw[192 x 192]   (tiny, L2-resident)
__global__ __launch_bounds__(256) void mm_plain(
    const float* __restrict__ X, const float* __restrict__ Mw,
    float* __restrict__ out)
{
    int g = blockIdx.x * 256 + threadIdx.x;      // 512*192 = 98304 threads
    int r = g / 192, c = g % 192;
    const float* xr = X + r * 192;
    float acc = 0.0f;
    #pragma unroll 8
    for (int t = 0; t < 192; ++t)
        acc = fmaf(xr[t], Mw[t * 192 + c], acc);
    out[g] = acc;
}

// y = (t0 [+ t1 [+ t2]]) / cnt ; out = (y - mean)/sqrt(var + 1e-5)
// mean/var over the 6144 elements of each n (LayerNorm over last two dims).
__global__ __launch_bounds__(256) void combine_ln(
    const float* __restrict__ t0, const float* __restrict__ t1,
    const float* __restrict__ t2, int cnt, float* __restrict__ out)
{
    const int nb   = blockIdx.x;     // 0..15
    const int tid  = threadIdx.x;
    const int base = nb * 6144;
    const float inv = 1.0f / (float)cnt;

    float y[24];
    float s = 0.0f, s2 = 0.0f;
    #pragma unroll
    for (int j = 0; j < 24; ++j) {
        int i = base + tid + j * 256;
        float v = t0[i];
        if (cnt > 1) v += t1[i];
        if (cnt > 2) v += t2[i];
        v *= inv;
        y[j] = v;
        s += v; s2 += v * v;
    }

    __shared__ float r1[256], r2[256];
    r1[tid] = s; r2[tid] = s2;
    __syncthreads();
    for (int st = 128; st > 0; st >>= 1) {
        if (tid < st) { r1[tid] += r1[tid + st]; r2[tid] += r2[tid + st]; }
        __syncthreads();
    }
    __shared__ float mean_s, rstd_s;
    if (tid == 0) {
        float mean = r1[0] * (1.0f / 6144.0f);
        float var  = r2[0] * (1.0f / 6144.0f) - mean * mean;
        mean_s = mean;
        rstd_s = rsqrtf(var + 1e-5f);
    }
    __syncthreads();
    float mean = mean_s, rstd = rstd_s;
    #pragma unroll
    for (int j = 0; j < 24; ++j)
        out[base + tid + j * 256] = (y[j] - mean) * rstd;
}

extern "C" void kernel_launch(void* const* d_in, const int* in_sizes, int n_in,
                              void* d_out, int out_size, void* d_ws, size_t ws_size,
                              hipStream_t stream)
{
    (void)in_sizes; (void)n_in; (void)out_size; (void)ws_size;
    const float* x   = (const float*)d_in[0];
    const float* s2t = (const float*)d_in[1];
    const float* t2s = (const float*)d_in[2];
    const float* s0  = (const float*)d_in[3];
    const float* s1  = (const float*)d_in[4];
    // d_in[5] (s2) unused for NUM_LAYERS = 4
    const float* t0  = (const float*)d_in[6];
    const float* t1w = (const float*)d_in[7];
    const float* t2w = (const float*)d_in[8];
    float* out = (float*)d_out;

    const size_t BUF = 16 * 6144;                 // 98304 floats per buffer
    float* ws  = (float*)d_ws;
    float* T1  = ws + 0 * BUF;   // layer1 pre-LN
    float* P3A = ws + 1 * BUF;   // layer3 j=0 term (from x, scale t2)
    float* H1  = ws + 2 * BUF;
    float* SS  = ws + 3 * BUF;   // x @ s1
    float* TS  = ws + 4 * BUF;   // t2s einsum term
    float* H2  = ws + 5 * BUF;
    float* TT  = ws + 6 * BUF;   // h1 @ t1
    float* P3C = ws + 7 * BUF;   // layer3 j=2 term (from h2, scale t0)

    dim3 gGemm(384), bGemm(256);
    dim3 gMM(384),   bMM(256);
    dim3 gLN(16),    bLN(256);

    // Pass 1: one s2t sweep (RT loads -> s2t lands in 192MB L2),
    //         two GEMMs sharing streams: layer1 (scale t0) + layer3-j0 (scale t2).
    gemm_scaled<true,  false><<<gGemm, bGemm, 0, stream>>>(x, s2t, t0, t2w, T1, P3A);
    combine_ln<<<gLN, bLN, 0, stream>>>(T1, nullptr, nullptr, 1, H1);

    // Layer 2: x@s1  +  t2s einsum(h1, scale s0); t2s streamed NON-TEMPORAL so
    // it does not evict the L2-resident s2t needed again in pass 3.
    mm_plain<<<gMM, bMM, 0, stream>>>(x, s1, SS);
    gemm_scaled<false, true ><<<gGemm, bGemm, 0, stream>>>(H1, t2s, s0, nullptr, TS, nullptr);
    combine_ln<<<gLN, bLN, 0, stream>>>(SS, TS, nullptr, 2, H2);

    // Layer 3: P3A + h1@t1 + s2t einsum(h2, scale t0); s2t mostly L2 hits now.
    mm_plain<<<gMM, bMM, 0, stream>>>(H1, t1w, TT);
    gemm_scaled<false, false><<<gGemm, bGemm, 0, stream>>>(H2, s2t, t0, nullptr, P3C, nullptr);
    combine_ln<<<gLN, bLN, 0, stream>>>(P3A, TT, P3C, 3, out);
}